// Selector_51264729645713
// MI455X (gfx1250) — compile-verified
//
#include <hip/hip_runtime.h>
#include <hip/hip_bf16.h>

typedef __attribute__((ext_vector_type(2))) float v2f;
typedef __attribute__((ext_vector_type(8))) float v8f;

#define B_  8
#define S_  128
#define V_  50000
#define L_  128
#define CL_ 512
#define QL_ 64
#define M_  (B_*S_)   // 1024 rows
#define KDIM 1536     // H2+H2+D
#define KMU  1836     // H2+H2+D+E

// ---------------------------------------------------------------------------
// Kernel 1: mixture-gate estimator.  One block per (b,s) row.
// ---------------------------------------------------------------------------
__global__ __launch_bounds__(256)
void est_kernel(const float* __restrict__ ctx_q, const float* __restrict__ ctx_c,
                const float* __restrict__ dec,   const float* __restrict__ cq,
                const float* __restrict__ gumbel, const float* __restrict__ eps,
                const float* __restrict__ W_mu,  const float* __restrict__ W_sigma,
                const float* __restrict__ W_pi,  const float* __restrict__ b_pi,
                const int* __restrict__ num_step, float* __restrict__ est)
{
    __shared__ float vs[KMU];
    __shared__ float hmu[L_];
    __shared__ float hsg[L_];
    __shared__ float hh[L_];
    __shared__ float pl[3];

    const int row = blockIdx.x;
    const int tid = threadIdx.x;

    for (int i = tid; i < KMU; i += 256) {
        float x;
        if (i < 512)       x = ctx_q[row*512 + i];
        else if (i < 1024) x = ctx_c[row*512 + (i - 512)];
        else if (i < 1536) x = dec  [row*512 + (i - 1024)];
        else               x = cq   [row*300 + (i - 1536)];
        vs[i] = x;
    }
    __syncthreads();

    if (tid < 128) {
        float acc = 0.f;
        for (int k = 0; k < KMU; ++k) acc = fmaf(vs[k], W_mu[k*L_ + tid], acc);
        hmu[tid] = tanhf(acc);
    } else {
        const int l = tid - 128;
        float acc = 0.f;
        for (int k = 0; k < KMU; ++k) acc = fmaf(vs[k], W_sigma[k*L_ + l], acc);
        hsg[l] = __expf(tanhf(acc));
    }
    __syncthreads();
    if (tid < 128) hh[tid] = hmu[tid] + hsg[tid] * eps[row*L_ + tid];
    __syncthreads();
    if (tid < 3) {
        float p = b_pi[tid];
        for (int l = 0; l < L_; ++l) p = fmaf(hh[l], W_pi[l*3 + tid], p);
        pl[tid] = p;
    }
    __syncthreads();
    if (tid == 0) {
        const float temp = 1.0f / (float)(*num_step);   // TEMP_INIT / num_step
        float m = fmaxf(pl[0], fmaxf(pl[1], pl[2]));
        float lse = logf(__expf(pl[0]-m) + __expf(pl[1]-m) + __expf(pl[2]-m));
        float z[3];
        #pragma unroll
        for (int k = 0; k < 3; ++k)
            z[k] = (gumbel[row*3 + k] + (pl[k] - m - lse)) / temp;
        float m2 = fmaxf(z[0], fmaxf(z[1], z[2]));
        float e0 = __expf(z[0]-m2), e1 = __expf(z[1]-m2), e2 = __expf(z[2]-m2);
        float inv = 1.0f / (e0 + e1 + e2);
        est[row*3+0] = e0*inv;
        est[row*3+1] = e1*inv;
        est[row*3+2] = e2*inv;
    }
}

// ---------------------------------------------------------------------------
// Kernel 2: vocab logits via f32 WMMA.  [1024 x 1536] @ [1536 x 50000] -> d_out
// Workgroup tile 64M x 128N, 8 waves, wave = 16M x 64N (4 accumulators,
// A fragment reused 4x).  B is stored TRANSPOSED in LDS so every fragment
// (A and B) is a single ds_load_b64 of two consecutive K elements.
// ---------------------------------------------------------------------------
#define KC  64
#define LDA 66     // padded strides: 66 words -> 264B, keeps b64 alignment and
#define LDB 66     // spreads rows across banks (bank = 2*row + k, conflict-free)
#define NT  128    // workgroup N tile

__global__ __launch_bounds__(256)
void gemm_kernel(const float* __restrict__ Aq, const float* __restrict__ Ac,
                 const float* __restrict__ Ad, const float* __restrict__ Wv,
                 const float* __restrict__ bias, float* __restrict__ out)
{
    __shared__ float sA [64 * LDA];   // sA [m][k]
    __shared__ float sBt[NT * LDB];   // sBt[n][k]  (transposed)

    const int tid    = threadIdx.x;
    const int m_base = blockIdx.x * 64;          // 16 blocks * 64 = 1024 rows exactly
    const int n0     = blockIdx.y * NT;          // guarded against V_
    const int wave   = tid >> 5;
    const int lane   = tid & 31;
    const int m_off  = (wave >> 1) * 16;
    const int n_off  = (wave & 1) * 64;
    const int fm     = lane & 15;                // fragment row/col within 16
    const int kh     = lane >> 4;                // K-half select (K pair 2*kh)

    v8f acc[4] = {{}, {}, {}, {}};

    for (int k0 = 0; k0 < KDIM; k0 += KC) {
        // A chunk never straddles the three concatenated tensors (512 % 64 == 0).
        const float* Asrc = (k0 < 512) ? Aq : (k0 < 1024) ? Ac : Ad;
        const int kk0 = k0 & 511;

        #pragma unroll
        for (int j = 0; j < 16; ++j) {           // 64x64 A tile, coalesced in K
            int i  = tid + 256*j;
            int ml = i >> 6, kl = i & 63;
            sA[ml*LDA + kl] = Asrc[(m_base + ml)*512 + kk0 + kl];
        }
        #pragma unroll
        for (int j = 0; j < 32; ++j) {           // 64k x 128n B tile, coalesced in N,
            int i  = tid + 256*j;                // stored transposed [n][k]
            int kl = i >> 7, nl = i & 127;
            int n  = n0 + nl;
            sBt[nl*LDB + kl] = (n < V_) ? Wv[(size_t)(k0 + kl)*V_ + n] : 0.f;
        }
        __syncthreads();

        #pragma unroll
        for (int kb = 0; kb < KC; kb += 4) {
            const int kf = kb + 2*kh;
            v2f a;
            a.x = sA[(m_off + fm)*LDA + kf];
            a.y = sA[(m_off + fm)*LDA + kf + 1];
            #pragma unroll
            for (int t = 0; t < 4; ++t) {
                v2f b;
                b.x = sBt[(n_off + 16*t + fm)*LDB + kf];
                b.y = sBt[(n_off + 16*t + fm)*LDB + kf + 1];
                acc[t] = __builtin_amdgcn_wmma_f32_16x16x4_f32(
                             false, a, false, b, (short)0, acc[t], false, false);
            }
        }
        __syncthreads();
    }

    // C/D layout: VGPR r -> M = r + 8*(lane>=16), N = lane%16
    const int gm = m_base + m_off + 8*kh;
    #pragma unroll
    for (int t = 0; t < 4; ++t) {
        const int gn = n0 + n_off + 16*t + fm;
        if (gn < V_) {
            const float bv = bias[gn];
            #pragma unroll
            for (int r = 0; r < 8; ++r)
                out[(size_t)(gm + r) * V_ + gn] = acc[t][r] + bv;
        }
    }
}

// ---------------------------------------------------------------------------
// Kernel 3: online-softmax row stats; folds estimator[...,2]/sum into scale.
// ---------------------------------------------------------------------------
__global__ __launch_bounds__(256)
void reduce_kernel(const float* __restrict__ logits, const float* __restrict__ est,
                   float* __restrict__ stats)
{
    __shared__ float sm[256], ss[256];
    const int row = blockIdx.x;
    const int tid = threadIdx.x;
    const float* p = logits + (size_t)row * V_;

    float m = -3.4e38f, s = 0.f;
    for (int v = tid; v < V_; v += 256) {
        float x = p[v];
        if (x > m) { s = s * __expf(m - x) + 1.0f; m = x; }
        else       { s += __expf(x - m); }
    }
    sm[tid] = m; ss[tid] = s;
    __syncthreads();
    for (int off = 128; off > 0; off >>= 1) {
        if (tid < off) {
            float m2 = sm[tid+off], s2 = ss[tid+off];
            float M  = fmaxf(sm[tid], m2);
            ss[tid]  = ss[tid]*__expf(sm[tid]-M) + s2*__expf(m2-M);
            sm[tid]  = M;
        }
        __syncthreads();
    }
    if (tid == 0) {
        stats[row*2+0] = sm[0];
        stats[row*2+1] = est[row*3+2] / ss[0];   // e2 / sum(exp)
    }
}

// ---------------------------------------------------------------------------
// Kernel 4: in-place  out = exp(logit - max) * (e2/sum),  float4 vectorized.
// ---------------------------------------------------------------------------
__global__ __launch_bounds__(256)
void scale_kernel(float* __restrict__ out, const float* __restrict__ stats)
{
    const int row = blockIdx.y;
    const int i4  = blockIdx.x * 256 + threadIdx.x;   // V_/4 = 12500 float4s
    if (i4 >= V_/4) return;
    const float m  = stats[row*2+0];
    const float sc = stats[row*2+1];
    float4* p = reinterpret_cast<float4*>(out + (size_t)row * V_) + i4;
    float4 x = *p;
    x.x = __expf(x.x - m) * sc;
    x.y = __expf(x.y - m) * sc;
    x.z = __expf(x.z - m) * sc;
    x.w = __expf(x.w - m) * sc;
    *p = x;
}

// ---------------------------------------------------------------------------
// Kernel 5: segment-sum scatter  out[b,s,idx[b,j]] += est[b,s,comp]*att[b,j,s]
// ---------------------------------------------------------------------------
__global__ __launch_bounds__(256)
void scatter_kernel(const float* __restrict__ att, const int* __restrict__ idx,
                    const float* __restrict__ est, float* __restrict__ out,
                    int src_len, int comp)
{
    const int g = blockIdx.x * 256 + threadIdx.x;
    const int total = B_ * src_len * S_;
    if (g >= total) return;
    const int s = g % S_;
    const int j = (g / S_) % src_len;
    const int b = g / (S_ * src_len);
    const float w = est[(b*S_ + s)*3 + comp];
    const int   v = idx[b*src_len + j];
    atomicAdd(out + (size_t)(b*S_ + s)*V_ + v, w * att[g]);
}

// ---------------------------------------------------------------------------
extern "C" void kernel_launch(void* const* d_in, const int* in_sizes, int n_in,
                              void* d_out, int out_size, void* d_ws, size_t ws_size,
                              hipStream_t stream)
{
    const float* context_q    = (const float*)d_in[0];
    const float* context_c    = (const float*)d_in[1];
    const float* decoded      = (const float*)d_in[2];
    const float* cq           = (const float*)d_in[3];
    const float* att_c        = (const float*)d_in[4];
    const float* att_q        = (const float*)d_in[5];
    const float* gumbel       = (const float*)d_in[6];
    const float* epsilon      = (const float*)d_in[7];
    const float* W_vocab      = (const float*)d_in[8];
    const float* b_vocab      = (const float*)d_in[9];
    const float* W_mu         = (const float*)d_in[10];
    const float* W_sigma      = (const float*)d_in[11];
    const float* W_pi         = (const float*)d_in[12];
    const float* b_pi         = (const float*)d_in[13];
    const int*   case_idx     = (const int*)d_in[14];
    const int*   question_idx = (const int*)d_in[15];
    const int*   num_step     = (const int*)d_in[16];

    float* out   = (float*)d_out;          // [1024, 50000] — doubles as logits buffer
    float* ws    = (float*)d_ws;
    float* est   = ws;                     // 1024*3 floats
    float* stats = ws + 4096;              // 1024*2 floats

    est_kernel<<<M_, 256, 0, stream>>>(context_q, context_c, decoded, cq,
                                       gumbel, epsilon, W_mu, W_sigma, W_pi, b_pi,
                                       num_step, est);

    dim3 gg(M_/64, (V_ + NT - 1)/NT);      // 16 x 391, blockIdx.x = M for L2 B-reuse
    gemm_kernel<<<gg, 256, 0, stream>>>(context_q, context_c, decoded,
                                        W_vocab, b_vocab, out);

    reduce_kernel<<<M_, 256, 0, stream>>>(out, est, stats);

    dim3 gs((V_/4 + 255)/256, M_);         // 49 x 1024
    scale_kernel<<<gs, 256, 0, stream>>>(out, stats);

    scatter_kernel<<<(B_*QL_*S_ + 255)/256, 256, 0, stream>>>(att_q, question_idx,
                                                              est, out, QL_, 0);
    scatter_kernel<<<(B_*CL_*S_ + 255)/256, 256, 0, stream>>>(att_c, case_idx,
                                                              est, out, CL_, 1);
}